// Alahi_Social_LSTM_44951127720421
// MI455X (gfx1250) — compile-verified
//
#include <hip/hip_runtime.h>
#include <hip/hip_bf16.h>
#include <math.h>

// ---------------- constants from the reference ----------------
#define NS    0.4f
#define GS    8
#define RNN   128
#define EMB   64
#define NMIX  20
#define NAG   1024
#define SOC   (GS*GS*RNN)      // 8192
#define G4    (4*RNN)          // 512
#define OUTD  (6*NMIX)         // 120

typedef __attribute__((ext_vector_type(2))) float v2f;
typedef __attribute__((ext_vector_type(4))) float v4f;
typedef __attribute__((ext_vector_type(8))) float v8f;
typedef __attribute__((ext_vector_type(4))) int   v4i;

// ---------------------------------------------------------------
// Kernel A: emb = relu(xoff @ W_emb.T + b_emb)  -> xin[:, 0:64]
// ---------------------------------------------------------------
__global__ void emb_kernel(const float* __restrict__ xoff,
                           const float* __restrict__ W_emb,
                           const float* __restrict__ b_emb,
                           float* __restrict__ xin)
{
    int idx = blockIdx.x * blockDim.x + threadIdx.x;  // NAG*EMB threads
    if (idx >= NAG * EMB) return;
    int n = idx >> 6;
    int e = idx & 63;
    float v = xoff[n*2 + 0] * W_emb[e*2 + 0]
            + xoff[n*2 + 1] * W_emb[e*2 + 1]
            + b_emb[e];
    xin[(size_t)n * (2*EMB) + e] = v > 0.f ? v : 0.f;
}

// ---------------------------------------------------------------
// Kernel B: social pooling scatter-add.
// One block per agent n; 128 threads = one h-channel each.
// LDS: 32 KB accumulator + 8 KB staged xabs (async-to-LDS if available).
// Fixed ascending-m accumulation order -> deterministic FP result.
// ---------------------------------------------------------------
__global__ void social_kernel(const float* __restrict__ xabs,
                              const float* __restrict__ h,     // h0[0] : (N, RNN)
                              float* __restrict__ social)      // (N, 8192)
{
    __shared__ float acc[GS*GS*RNN];   // 32 KB
    __shared__ float sx[NAG * 2];      // 8 KB staged xabs
    const int n = blockIdx.x;
    const int t = threadIdx.x;         // 0..127

    #pragma unroll
    for (int g = 0; g < GS*GS; ++g) acc[g*RNN + t] = 0.f;

    // ---- stage xabs (2048 floats) into LDS ----
#if defined(__gfx1250__) && __has_builtin(__builtin_amdgcn_global_load_async_to_lds_b128)
    // 4 async ops x 128 lanes x 16B = 8192 B, tracked by ASYNCcnt
    typedef __attribute__((address_space(1))) v4i* gp_v4i;
    typedef __attribute__((address_space(3))) v4i* lp_v4i;
    #pragma unroll
    for (int i = 0; i < 4; ++i) {
        int f4 = i * 128 + t;  // 16-byte chunk index
        __builtin_amdgcn_global_load_async_to_lds_b128(
            (gp_v4i)(xabs + 4*f4),
            (lp_v4i)(sx + 4*f4),
            0, 0);
    }
  #if __has_builtin(__builtin_amdgcn_s_wait_asynccnt)
    __builtin_amdgcn_s_wait_asynccnt(0);
  #else
    asm volatile("s_wait_asynccnt 0" ::: "memory");
  #endif
#else
    #pragma unroll
    for (int i = 0; i < 4; ++i) {
        int f4 = i * 128 + t;
        ((v4f*)sx)[f4] = ((const v4f*)xabs)[f4];
    }
#endif
    __syncthreads();

    const float xn = sx[n*2 + 0] - 0.5f * NS;
    const float yn = sx[n*2 + 1] - 0.5f * NS;
    const float inv = (float)GS / NS;  // 20

    for (int m = 0; m < NAG; ++m) {
        if (m == n) continue;
        float dx = sx[m*2 + 0] - xn;
        float dy = sx[m*2 + 1] - yn;
        if (dx >= 0.f && dx < NS && dy >= 0.f && dy < NS) {
            int cx = (int)floorf(dx * inv);
            int cy = (int)floorf(dy * inv);
            if (cx >= 0 && cx < GS && cy >= 0 && cy < GS) {
                int cell = cx + cy * GS;
                acc[cell*RNN + t] += h[(size_t)m * RNN + t];
            }
        }
    }
    __syncthreads();

    float* dst = social + (size_t)n * SOC;
    #pragma unroll
    for (int g = 0; g < GS*GS; ++g) dst[g*RNN + t] = acc[g*RNN + t];
}

// ---------------------------------------------------------------
// Kernel C: pool = relu(social @ W_soc.T + b_soc) -> xin[:, 64:128]
// FP32 WMMA, one wave per 16x64 output strip (4 accumulators):
// per K-step: 1 A load + 4 B loads + 4 WMMAs (A reused across e-tiles).
// ---------------------------------------------------------------
__global__ void pool_wmma_kernel(const float* __restrict__ social, // (N, 8192)
                                 const float* __restrict__ W_soc,  // (64, 8192)
                                 const float* __restrict__ b_soc,  // (64)
                                 float* __restrict__ xin)          // (N, 128)
{
    const int lane = threadIdx.x & 31;
    const int wave = threadIdx.x >> 5;
    const int mt   = blockIdx.x * (blockDim.x >> 5) + wave;  // 0..63 m-tiles
    const int m0   = mt * 16;

    const int row = lane & 15;
    const int kb  = (lane >> 4) << 1;  // 0 or 2

    const float* aP = social + (size_t)(m0 + row) * SOC + kb;
    const float* bP[4];
    #pragma unroll
    for (int te = 0; te < 4; ++te)
        bP[te] = W_soc + (size_t)(te*16 + row) * SOC + kb;

    v8f c[4] = {};
    #pragma unroll 2
    for (int k = 0; k < SOC; k += 4) {
        v2f a = *(const v2f*)(aP + k);
        #pragma unroll
        for (int te = 0; te < 4; ++te) {
            v2f b = *(const v2f*)(bP[te] + k);
            c[te] = __builtin_amdgcn_wmma_f32_16x16x4_f32(
                        false, a, false, b, (short)0, c[te], false, false);
        }
    }

    // C/D layout: VGPR v holds (M = v + 8*(lane>>4), N = lane&15)
    const int col   = lane & 15;
    const int mbase = m0 + ((lane >> 4) << 3);
    #pragma unroll
    for (int te = 0; te < 4; ++te) {
        const int e = te*16 + col;
        const float bias = b_soc[e];
        #pragma unroll
        for (int v = 0; v < 8; ++v) {
            float val = c[te][v] + bias;
            val = val > 0.f ? val : 0.f;
            xin[(size_t)(mbase + v) * (2*EMB) + EMB + e] = val;
        }
    }
}

// ---------------------------------------------------------------
// Kernel D: gates = xin @ W_ih.T + h @ W_hh.T + b_ih + b_hh
// One wave per 16x64 strip of the (1024,512) gates; two K=128 loops
// accumulated into 4 shared accumulators.
// ---------------------------------------------------------------
__global__ void gates_wmma_kernel(const float* __restrict__ xin,   // (N, 128)
                                  const float* __restrict__ h,     // (N, 128)
                                  const float* __restrict__ W_ih,  // (512, 128)
                                  const float* __restrict__ W_hh,  // (512, 128)
                                  const float* __restrict__ b_ih,  // (512)
                                  const float* __restrict__ b_hh,  // (512)
                                  float* __restrict__ gates)       // (N, 512)
{
    const int lane  = threadIdx.x & 31;
    const int wave  = threadIdx.x >> 5;
    const int strip = blockIdx.x * (blockDim.x >> 5) + wave;  // 512 strips
    const int strips_j = G4 / 64;                             // 8
    const int m0 = (strip / strips_j) * 16;                   // 0..1008
    const int j0 = (strip % strips_j) * 64;                   // 0..448

    const int row = lane & 15;
    const int kb  = (lane >> 4) << 1;

    v8f c[4] = {};
    {
        const float* aP = xin + (size_t)(m0 + row) * (2*EMB) + kb;
        #pragma unroll
        for (int k = 0; k < 2*EMB; k += 4) {
            v2f a = *(const v2f*)(aP + k);
            #pragma unroll
            for (int tj = 0; tj < 4; ++tj) {
                v2f b = *(const v2f*)(W_ih + (size_t)(j0 + tj*16 + row) * (2*EMB) + kb + k);
                c[tj] = __builtin_amdgcn_wmma_f32_16x16x4_f32(
                            false, a, false, b, (short)0, c[tj], false, false);
            }
        }
    }
    {
        const float* aP = h + (size_t)(m0 + row) * RNN + kb;
        #pragma unroll
        for (int k = 0; k < RNN; k += 4) {
            v2f a = *(const v2f*)(aP + k);
            #pragma unroll
            for (int tj = 0; tj < 4; ++tj) {
                v2f b = *(const v2f*)(W_hh + (size_t)(j0 + tj*16 + row) * RNN + kb + k);
                c[tj] = __builtin_amdgcn_wmma_f32_16x16x4_f32(
                            false, a, false, b, (short)0, c[tj], false, false);
            }
        }
    }

    const int col   = lane & 15;
    const int mbase = m0 + ((lane >> 4) << 3);
    #pragma unroll
    for (int tj = 0; tj < 4; ++tj) {
        const int j = j0 + tj*16 + col;
        const float bias = b_ih[j] + b_hh[j];
        #pragma unroll
        for (int v = 0; v < 8; ++v) {
            gates[(size_t)(mbase + v) * G4 + j] = c[tj][v] + bias;
        }
    }
}

// ---------------------------------------------------------------
// Kernel E: LSTM pointwise.
// ---------------------------------------------------------------
__device__ __forceinline__ float sigm(float x) { return 1.f / (1.f + expf(-x)); }

__global__ void lstm_kernel(const float* __restrict__ gates,  // (N, 512)
                            const float* __restrict__ c0,     // (N, 128)
                            float* __restrict__ hn)           // (N, 128)
{
    int idx = blockIdx.x * blockDim.x + threadIdx.x;  // NAG*RNN
    if (idx >= NAG * RNN) return;
    int n = idx >> 7;
    int r = idx & 127;
    const float* gr = gates + (size_t)n * G4;
    float gi = gr[r];
    float gf = gr[RNN + r];
    float gg = gr[2*RNN + r];
    float go = gr[3*RNN + r];
    float cc = sigm(gf) * c0[idx] + sigm(gi) * tanhf(gg);
    hn[idx]  = sigm(go) * tanhf(cc);
}

// ---------------------------------------------------------------
// Kernel F: out = hn @ W_out.T + b_out, scattered into the
// 6-way split order (mu1|mu2|ls1|ls2|rho|pi), each (N, 20).
// ---------------------------------------------------------------
__global__ void out_kernel(const float* __restrict__ hn,     // (N, 128)
                           const float* __restrict__ W_out,  // (120, 128)
                           const float* __restrict__ b_out,  // (120)
                           float* __restrict__ out)          // 6 * N * 20
{
    int idx = blockIdx.x * blockDim.x + threadIdx.x;  // NAG*OUTD
    if (idx >= NAG * OUTD) return;
    int n = idx / OUTD;
    int j = idx % OUTD;
    const float* hr = hn    + (size_t)n * RNN;
    const float* wr = W_out + (size_t)j * RNN;
    float s = b_out[j];
    #pragma unroll 8
    for (int k = 0; k < RNN; ++k) s += hr[k] * wr[k];
    int grp = j / NMIX;
    int col = j % NMIX;
    out[(size_t)grp * NAG * NMIX + (size_t)n * NMIX + col] = s;
}

// ---------------------------------------------------------------
extern "C" void kernel_launch(void* const* d_in, const int* in_sizes, int n_in,
                              void* d_out, int out_size, void* d_ws, size_t ws_size,
                              hipStream_t stream)
{
    const float* xoff  = (const float*)d_in[0];
    const float* xabs  = (const float*)d_in[1];
    const float* h0    = (const float*)d_in[2];   // (1, N, 128) -> flat (N,128)
    const float* c0    = (const float*)d_in[3];
    const float* W_emb = (const float*)d_in[4];
    const float* b_emb = (const float*)d_in[5];
    const float* W_soc = (const float*)d_in[6];
    const float* b_soc = (const float*)d_in[7];
    const float* W_ih  = (const float*)d_in[8];
    const float* W_hh  = (const float*)d_in[9];
    const float* b_ih  = (const float*)d_in[10];
    const float* b_hh  = (const float*)d_in[11];
    const float* W_out = (const float*)d_in[12];
    const float* b_out = (const float*)d_in[13];
    float* out = (float*)d_out;

    // workspace layout (f32 elements)
    float* ws     = (float*)d_ws;
    float* xin    = ws;                                   // N*128
    float* social = xin    + (size_t)NAG * (2*EMB);       // N*8192
    float* gates  = social + (size_t)NAG * SOC;           // N*512
    float* hn     = gates  + (size_t)NAG * G4;            // N*128
    (void)ws_size; (void)in_sizes; (void)n_in; (void)out_size;

    // A: emb -> xin[:, :64]
    emb_kernel<<<(NAG*EMB + 255) / 256, 256, 0, stream>>>(xoff, W_emb, b_emb, xin);

    // B: social scatter-add (one block per agent)
    social_kernel<<<NAG, RNN, 0, stream>>>(xabs, h0, social);

    // C: pool GEMM (1024x8192x64), 64 strip-waves -> 16 blocks x 128
    pool_wmma_kernel<<<16, 128, 0, stream>>>(social, W_soc, b_soc, xin);

    // D: gates GEMM (1024x128x512 twice), 512 strip-waves -> 128 blocks x 128
    gates_wmma_kernel<<<128, 128, 0, stream>>>(xin, h0, W_ih, W_hh, b_ih, b_hh, gates);

    // E: LSTM pointwise
    lstm_kernel<<<(NAG*RNN + 255) / 256, 256, 0, stream>>>(gates, c0, hn);

    // F: output GEMV + 6-way split scatter
    out_kernel<<<(NAG*OUTD + 255) / 256, 256, 0, stream>>>(hn, W_out, b_out, out);
}